// EvolvingMultiHeadAttention_65704409694231
// MI455X (gfx1250) — compile-verified
//
#include <hip/hip_runtime.h>
#include <hip/hip_bf16.h>
#include <math.h>
#include <stdint.h>

#define S_LEN 2048
#define H_DIM 1024
#define N_HEADS 16
#define HEAD_DIM 64
#define SH ((size_t)S_LEN * H_DIM)
#define EV_STRIDE (S_LEN + 4)          // padded LDS row stride (floats)

typedef __attribute__((ext_vector_type(2))) float v2f;
typedef __attribute__((ext_vector_type(8))) float v8f;

// ---------------------------------------------------------------------------
// Generic 2048x1024 @ (1024x1024)^T GEMM via V_WMMA_F32_16X16X4_F32.
// One wave per 16x16 output tile; 8 waves per 256-thread block share the
// same 16-row A strip (consecutive tn) for L1 reuse.
// dst[s][n] = sum_i A[s][i] * W[n][i]
// ---------------------------------------------------------------------------
__global__ __launch_bounds__(256) void gemm_xwt(const float* __restrict__ A,
                                                const float* __restrict__ W,
                                                float* __restrict__ dst) {
  const int w    = threadIdx.x >> 5;
  const int lane = threadIdx.x & 31;
  const int half = lane >> 4;
  const int m    = lane & 15;
  const int tile = blockIdx.x * 8 + w;        // 0..8191
  const int tm = tile >> 6;                   // 0..127 (rows of 16)
  const int tn = tile & 63;                   // 0..63  (cols of 16)
  const int mbase = tm * 16, nbase = tn * 16;

  const float* arow = A + (size_t)(mbase + m) * H_DIM + 2 * half;
  const float* brow = W + (size_t)(nbase + m) * H_DIM + 2 * half;

  v8f c = {};
  for (int kk = 0; kk < H_DIM; kk += 4) {
    v2f a = *(const v2f*)(arow + kk);
    v2f b = *(const v2f*)(brow + kk);
    c = __builtin_amdgcn_wmma_f32_16x16x4_f32(false, a, false, b, (short)0, c,
                                              false, false);
  }
  float* o = dst + (size_t)(mbase + 8 * half) * H_DIM + nbase + m;
#pragma unroll
  for (int r = 0; r < 8; ++r) o[(size_t)r * H_DIM] = c[r];
}

// ---------------------------------------------------------------------------
// temporal = hs @ W_temp.T + b_temp ; gate = sigmoid(hs @ W_mem.T + b_mem)
// Stored as [s][head] (S x 16). One thread per (s, head).
// ---------------------------------------------------------------------------
__global__ __launch_bounds__(256) void temp_gate_kernel(
    const float* __restrict__ hs, const float* __restrict__ Wt,
    const float* __restrict__ bt, const float* __restrict__ Wm,
    const float* __restrict__ bm, float* __restrict__ tout,
    float* __restrict__ gout) {
  int t = blockIdx.x * blockDim.x + threadIdx.x;
  if (t >= S_LEN * N_HEADS) return;
  int s = t >> 4, hh = t & 15;
  const float* x  = hs + (size_t)s * H_DIM;
  const float* wt = Wt + (size_t)hh * H_DIM;
  const float* wm = Wm + (size_t)hh * H_DIM;
  float s1 = 0.f, s2 = 0.f;
  for (int i = 0; i < H_DIM; ++i) {
    float xv = x[i];
    s1 += xv * wt[i];
    s2 += xv * wm[i];
  }
  tout[t] = s1 + bt[hh];
  gout[t] = 1.0f / (1.0f + __expf(-(s2 + bm[hh])));
}

// ---------------------------------------------------------------------------
// Fused attention: one block = (head h, 16-row q tile).
// Pass A : scores (WMMA f32 16x16x4) + evolution blend -> 16x2048 tile in LDS,
//          single read of attention_memory (prefetched one tile ahead),
//          running row max.
// Pass B : exp(e - max) in LDS, row sums via ds_add_f32 atomics.
// Pass C1: normalize in LDS.
// Async  : weights tile streamed LDS -> HBM with
//          GLOBAL_STORE_ASYNC_FROM_LDS_B128, overlapped with...
// Pass C2: ctx = weights @ v with WMMA, A operand straight from LDS.
// ---------------------------------------------------------------------------
__global__ __launch_bounds__(256) void attn_kernel(
    const float* __restrict__ q, const float* __restrict__ kmat,
    const float* __restrict__ vmat, const float* __restrict__ mask,
    const float* __restrict__ mem, const float* __restrict__ temporal,
    const float* __restrict__ gate, const float* __restrict__ evo,
    const float* __restrict__ dec, float* __restrict__ weights_out,
    float* __restrict__ ctx_out) {
  __shared__ float s_ev[16][EV_STRIDE];   // +4 pad: conflict-free WMMA-A loads
  __shared__ float s_pmax[8][16];
  __shared__ float s_rowmax[16];
  __shared__ float s_rowsum[16];
  __shared__ float s_ctx[16][HEAD_DIM];

  const int tid  = threadIdx.x;
  const int w    = tid >> 5;
  const int lane = tid & 31;
  const int half = lane >> 4;
  const int m    = lane & 15;
  const int h     = blockIdx.x >> 7;   // head
  const int qt    = blockIdx.x & 127;  // q tile
  const int qbase = qt * 16;
  const int hbase = h * HEAD_DIM;

  const float ef  = evo[h];
  const float md  = dec[h];
  const float omf = 1.0f - ef;
  const float scale = 0.125f;  // 1/sqrt(64)

  for (int i = tid; i < 16 * HEAD_DIM; i += 256) s_ctx[i >> 6][i & 63] = 0.f;
  if (tid < 16) s_rowsum[tid] = 0.f;

  // q fragment for this lane's (m, half) slot, register-resident.
  v2f qa[16];
  {
    const float* qrow = q + (size_t)(qbase + m) * H_DIM + hbase + 2 * half;
#pragma unroll
    for (int t = 0; t < 16; ++t) qa[t] = *(const v2f*)(qrow + 4 * t);
  }
  // Per-row fused coefficients for the 8 D-rows this lane produces.
  float tarr[8], garr[8];
#pragma unroll
  for (int r = 0; r < 8; ++r) {
    int row = qbase + r + 8 * half;
    tarr[r] = temporal[row * N_HEADS + h] * ef * 0.1f;
    garr[r] = gate[row * N_HEADS + h] * ef * md;
  }

  float rmax[8];
#pragma unroll
  for (int r = 0; r < 8; ++r) rmax[r] = -3.0e38f;

  // Per-lane base pointer into the attention_memory stream (HBM-resident).
  const float* membase =
      mem + ((size_t)h * S_LEN + qbase + 8 * half) * S_LEN + m;

  // ----- Pass A: this wave covers key cols [w*256, w*256+256) -----
  for (int ktl = 0; ktl < 16; ++ktl) {
    const int kbase = (w * 16 + ktl) * 16;

    // Prefetch next tile's attention_memory rows (hide HBM latency behind
    // the 16-deep WMMA chain). Emits global_prefetch_b8.
    if (ktl < 15) {
      const float* p = membase + kbase + 16;
#pragma unroll
      for (int r = 0; r < 8; ++r)
        __builtin_prefetch(p + (size_t)r * S_LEN, 0, 3);
    }

    const float* krow =
        kmat + (size_t)(kbase + m) * H_DIM + hbase + 2 * half;
    v8f c = {};
#pragma unroll
    for (int t = 0; t < 16; ++t) {
      v2f b = *(const v2f*)(krow + 4 * t);
      c = __builtin_amdgcn_wmma_f32_16x16x4_f32(false, qa[t], false, b,
                                                (short)0, c, false, false);
    }
    const int col = kbase + m;
#pragma unroll
    for (int r = 0; r < 8; ++r) {
      const int row = qbase + r + 8 * half;
      float e = (c[r] * scale + mask[(size_t)row * S_LEN + col]) * omf +
                membase[(size_t)r * S_LEN + kbase] * garr[r] +
                tarr[r];
      s_ev[r + 8 * half][col] = e;
      rmax[r] = fmaxf(rmax[r], e);
    }
  }
  // reduce max across the 16 lanes of each half (xor 1,2,4,8 stays in-half)
#pragma unroll
  for (int r = 0; r < 8; ++r) {
    float v = rmax[r];
    for (int off = 8; off >= 1; off >>= 1) v = fmaxf(v, __shfl_xor(v, off, 32));
    rmax[r] = v;
  }
  if (m == 0) {
#pragma unroll
    for (int r = 0; r < 8; ++r) s_pmax[w][r + 8 * half] = rmax[r];
  }
  __syncthreads();
  if (tid < 16) {
    float mx = -3.0e38f;
    for (int ww = 0; ww < 8; ++ww) mx = fmaxf(mx, s_pmax[ww][tid]);
    s_rowmax[tid] = mx;
  }
  __syncthreads();

  // ----- Pass B: exp + row sums (16 threads per row) -----
  {
    const int row = tid >> 4, c16 = tid & 15;
    const float mx = s_rowmax[row];
    float sum = 0.f;
    for (int col = c16; col < S_LEN; col += 16) {
      float p = __expf(s_ev[row][col] - mx);
      s_ev[row][col] = p;
      sum += p;
    }
    atomicAdd(&s_rowsum[row], sum);  // ds_add_f32
  }
  __syncthreads();
  if (tid < 16) s_rowsum[tid] = 1.0f / s_rowsum[tid];
  __syncthreads();

  // ----- Pass C1: normalize in LDS (global write handled by async DMA) -----
  for (int idx = tid; idx < 16 * S_LEN; idx += 256) {
    const int row = idx >> 11, col = idx & (S_LEN - 1);
    s_ev[row][col] *= s_rowsum[row];
  }
  __syncthreads();

  // ----- Async weights write-out: LDS -> HBM, overlapped with pass C2 -----
  // Destination block is one contiguous 128KB region:
  //   weights_out[(h*S + qbase + 0..15)*S + 0..2047]
  {
    const uint32_t lds_base = (uint32_t)(uintptr_t)(&s_ev[0][0]);
    const uint64_t wbase =
        (uint64_t)(uintptr_t)(weights_out + ((size_t)h * S_LEN + qbase) * S_LEN);
    for (int it = 0; it < 32; ++it) {
      const int chunk = it * 256 + tid;            // 16B chunk id, 0..8191
      const int row   = chunk >> 9;                // 512 chunks per row
      const int col4  = chunk & 511;
      const uint32_t lds_addr =
          lds_base + (uint32_t)(row * (EV_STRIDE * 4) + col4 * 16);
      const uint64_t gaddr = wbase + (uint64_t)chunk * 16u;
      asm volatile("global_store_async_from_lds_b128 %0, %1, off"
                   :
                   : "v"(gaddr), "v"(lds_addr)
                   : "memory");
    }
  }

  // ----- Pass C2: ctx = weights @ v, A from LDS -----
  {
    v8f c4[4] = {};
    const int kw = w * 256;
    for (int s4 = 0; s4 < 64; ++s4) {
      const int kk = kw + s4 * 4;
      v2f a;
      a.x = s_ev[m][kk + 2 * half];
      a.y = s_ev[m][kk + 2 * half + 1];
      const float* v0 = vmat + (size_t)(kk + 2 * half) * H_DIM + hbase + m;
#pragma unroll
      for (int nt = 0; nt < 4; ++nt) {
        v2f b;
        b.x = v0[nt * 16];
        b.y = v0[H_DIM + nt * 16];
        c4[nt] = __builtin_amdgcn_wmma_f32_16x16x4_f32(false, a, false, b,
                                                       (short)0, c4[nt],
                                                       false, false);
      }
    }
#pragma unroll
    for (int nt = 0; nt < 4; ++nt)
#pragma unroll
      for (int r = 0; r < 8; ++r)
        atomicAdd(&s_ctx[r + 8 * half][nt * 16 + m], c4[nt][r]);  // ds_add_f32
  }
  __syncthreads();
  for (int idx = tid; idx < 16 * HEAD_DIM; idx += 256) {
    const int row = idx >> 6, d = idx & 63;
    ctx_out[(size_t)(qbase + row) * H_DIM + hbase + d] = s_ctx[row][d];
  }

  // Ensure all async LDS->global stores drained before wave exit.
  asm volatile("s_wait_asynccnt 0x0" ::: "memory");
}

// ---------------------------------------------------------------------------
extern "C" void kernel_launch(void* const* d_in, const int* in_sizes, int n_in,
                              void* d_out, int out_size, void* d_ws,
                              size_t ws_size, hipStream_t stream) {
  (void)in_sizes; (void)n_in; (void)out_size; (void)ws_size;
  const float* hs   = (const float*)d_in[0];
  const float* mask = (const float*)d_in[1];
  const float* mem  = (const float*)d_in[2];
  const float* Wq   = (const float*)d_in[3];
  const float* Wk   = (const float*)d_in[4];
  const float* Wv   = (const float*)d_in[5];
  const float* Wo   = (const float*)d_in[6];
  const float* Wt   = (const float*)d_in[7];
  const float* bt   = (const float*)d_in[8];
  const float* Wm   = (const float*)d_in[9];
  const float* bm   = (const float*)d_in[10];
  const float* evo  = (const float*)d_in[11];
  const float* dec  = (const float*)d_in[12];

  float* out  = (float*)d_out;              // (S, H)
  float* wout = out + SH;                   // (NH, S, S) new_attention_memory

  float* ws   = (float*)d_ws;               // needs ~33.8 MB
  float* q    = ws;
  float* k    = q + SH;
  float* v    = k + SH;
  float* ctx  = v + SH;
  float* tbuf = ctx + SH;
  float* gbuf = tbuf + (size_t)S_LEN * N_HEADS;

  gemm_xwt<<<1024, 256, 0, stream>>>(hs, Wq, q);
  gemm_xwt<<<1024, 256, 0, stream>>>(hs, Wk, k);
  gemm_xwt<<<1024, 256, 0, stream>>>(hs, Wv, v);
  temp_gate_kernel<<<(S_LEN * N_HEADS + 255) / 256, 256, 0, stream>>>(
      hs, Wt, bt, Wm, bm, tbuf, gbuf);
  attn_kernel<<<N_HEADS * (S_LEN / 16), 256, 0, stream>>>(
      q, k, v, mask, mem, tbuf, gbuf, evo, dec, wout, ctx);
  gemm_xwt<<<1024, 256, 0, stream>>>(ctx, Wo, out);
}